// ChamferLoss_32186484916784
// MI455X (gfx1250) — compile-verified
//
#include <hip/hip_runtime.h>
#include <hip/hip_bf16.h>
#include <math.h>

typedef __attribute__((ext_vector_type(2))) float v2f;
typedef __attribute__((ext_vector_type(8))) float v8f;

#define TILE 16
#define WAVES_PER_BLOCK 8
#define BLOCK_THREADS (WAVES_PER_BLOCK * 32)
#define COL_CHUNKS 4

// Initialize rowmin arrays to +inf bit pattern (0x7F800000).
__global__ void chamfer_init_kernel(unsigned int* __restrict__ p, int n) {
    int i = blockIdx.x * blockDim.x + threadIdx.x;
    if (i < n) p[i] = 0x7F800000u;
}

// For each row x in X (16 rows per wave), compute min over a chunk of Y columns of
// ||x - y||^2 using V_WMMA_F32_16X16X4_F32:
//   A row m  = [-2*x0, -2*x1, -2*x2, 1]
//   B col n  = [ y0,    y1,    y2,   ||y||^2 ]
//   C[m][n]  = ||y||^2 - 2*(x . y)      ->  d2 = C[m][n] + ||x||^2
// Column loop is software-pipelined 2x: both global_load_b96's issue before either
// WMMA so load latency overlaps the matrix op + min epilogue.
__global__ __launch_bounds__(BLOCK_THREADS)
void chamfer_rowmin_kernel(const float* __restrict__ X, const float* __restrict__ Y,
                           int nX, int nY, unsigned int* __restrict__ rowmin) {
    const int lane  = threadIdx.x & 31;
    const int wave  = blockIdx.x * WAVES_PER_BLOCK + (threadIdx.x >> 5);
    const int rowTile = wave / COL_CHUNKS;
    const int chunk   = wave % COL_CHUNKS;
    const int row_base = rowTile * TILE;
    if (row_base >= nX) return;

    const int colsPerChunk = nY / COL_CHUNKS;
    const int col_begin = chunk * colsPerChunk;
    const int nTiles    = colsPerChunk / TILE;   // 8192/4/16 = 128, even

    const int hi = lane >> 4;        // 0: lanes 0-15 (K=0,1), 1: lanes 16-31 (K=2,3)
    const int lrow = row_base + (lane & 15);

    // A matrix: lane<16 -> {-2*x0, -2*x1}; lane>=16 -> {-2*x2, 1.0}
    const float ax0 = X[lrow * 3 + 0];
    const float ax1 = X[lrow * 3 + 1];
    const float ax2 = X[lrow * 3 + 2];
    v2f a;
    a.x = hi ? (-2.0f * ax2) : (-2.0f * ax0);
    a.y = hi ? 1.0f          : (-2.0f * ax1);

    // ||x||^2 for the row this lane's k-th C element belongs to:
    // C vgpr k, this lane -> row (row_base + k + 8*hi)
    float xsq[8];
#pragma unroll
    for (int k = 0; k < 8; ++k) {
        const int r = row_base + k + 8 * hi;
        const float u0 = X[r * 3 + 0];
        const float u1 = X[r * 3 + 1];
        const float u2 = X[r * 3 + 2];
        xsq[k] = fmaf(u0, u0, fmaf(u1, u1, u2 * u2));
    }

    float rmin[8];
#pragma unroll
    for (int k = 0; k < 8; ++k) rmin[k] = __builtin_inff();

    // Per-lane Y pointer: column (col_begin + lane&15), advanced 16 cols/tile.
    const float* __restrict__ yp = Y + (size_t)(col_begin + (lane & 15)) * 3;

    for (int t = 0; t < nTiles; t += 2) {
        // Issue both 12-byte loads first (two loads in flight).
        const float p0x = yp[0],  p0y = yp[1],  p0z = yp[2];
        const float p1x = yp[48], p1y = yp[49], p1z = yp[50];   // +16 cols * 3
        yp += 96;

        {
            const float ysq = fmaf(p0x, p0x, fmaf(p0y, p0y, p0z * p0z));
            v2f b;
            b.x = hi ? p0z : p0x;
            b.y = hi ? ysq : p0y;
            v8f c = {};
            c = __builtin_amdgcn_wmma_f32_16x16x4_f32(
                    false, a, false, b, (short)0, c, false, false);
#pragma unroll
            for (int k = 0; k < 8; ++k)
                rmin[k] = fminf(rmin[k], c[k] + xsq[k]);
        }
        {
            const float ysq = fmaf(p1x, p1x, fmaf(p1y, p1y, p1z * p1z));
            v2f b;
            b.x = hi ? p1z : p1x;
            b.y = hi ? ysq : p1y;
            v8f c = {};
            c = __builtin_amdgcn_wmma_f32_16x16x4_f32(
                    false, a, false, b, (short)0, c, false, false);
#pragma unroll
            for (int k = 0; k < 8; ++k)
                rmin[k] = fminf(rmin[k], c[k] + xsq[k]);
        }
    }

    // Min across the 16 lanes of each half (each half holds disjoint rows).
#pragma unroll
    for (int k = 0; k < 8; ++k) {
        float v = rmin[k];
        v = fminf(v, __shfl_xor(v, 1, 32));
        v = fminf(v, __shfl_xor(v, 2, 32));
        v = fminf(v, __shfl_xor(v, 4, 32));
        v = fminf(v, __shfl_xor(v, 8, 32));
        rmin[k] = v;
    }

    if ((lane & 15) == 0) {
#pragma unroll
        for (int k = 0; k < 8; ++k) {
            const unsigned int bits = __float_as_uint(fmaxf(rmin[k], 0.0f));
            atomicMin(&rowmin[row_base + k + 8 * hi], bits);
        }
    }
}

// Sum sqrt(min d2) over both directions, divide by counts, write scalar.
__global__ __launch_bounds__(256)
void chamfer_finalize_kernel(const unsigned int* __restrict__ rx,
                             const unsigned int* __restrict__ ry,
                             int N, int M, float* __restrict__ out) {
    __shared__ float smem[256];
    float accx = 0.0f;
    for (int i = threadIdx.x; i < N; i += 256)
        accx += sqrtf(__uint_as_float(rx[i]));
    float accy = 0.0f;
    for (int i = threadIdx.x; i < M; i += 256)
        accy += sqrtf(__uint_as_float(ry[i]));
    smem[threadIdx.x] = accx * (1.0f / (float)N) + accy * (1.0f / (float)M);
    __syncthreads();
    for (int s = 128; s > 0; s >>= 1) {
        if (threadIdx.x < s) smem[threadIdx.x] += smem[threadIdx.x + s];
        __syncthreads();
    }
    if (threadIdx.x == 0) out[0] = smem[0];
}

extern "C" void kernel_launch(void* const* d_in, const int* in_sizes, int n_in,
                              void* d_out, int out_size, void* d_ws, size_t ws_size,
                              hipStream_t stream) {
    const float* pred  = (const float*)d_in[0];
    const float* label = (const float*)d_in[1];
    const int N = in_sizes[0] / 3;   // 8192
    const int M = in_sizes[1] / 3;   // 8192

    unsigned int* rx = (unsigned int*)d_ws;      // [N] min ||x - y||^2 bits
    unsigned int* ry = rx + N;                   // [M] min ||y - x||^2 bits

    // Init mins to +inf.
    {
        const int total = N + M;
        chamfer_init_kernel<<<(total + 255) / 256, 256, 0, stream>>>(rx, total);
    }

    // Pass 1: rows = pred, cols = label.
    {
        const int rowTiles = (N + TILE - 1) / TILE;
        const int waves = rowTiles * COL_CHUNKS;
        const int blocks = (waves + WAVES_PER_BLOCK - 1) / WAVES_PER_BLOCK;
        chamfer_rowmin_kernel<<<blocks, BLOCK_THREADS, 0, stream>>>(
            pred, label, N, M, rx);
    }
    // Pass 2: rows = label, cols = pred.
    {
        const int rowTiles = (M + TILE - 1) / TILE;
        const int waves = rowTiles * COL_CHUNKS;
        const int blocks = (waves + WAVES_PER_BLOCK - 1) / WAVES_PER_BLOCK;
        chamfer_rowmin_kernel<<<blocks, BLOCK_THREADS, 0, stream>>>(
            label, pred, M, N, ry);
    }

    chamfer_finalize_kernel<<<1, 256, 0, stream>>>(rx, ry, N, M, (float*)d_out);
}